// System2Graph_89970974916667
// MI455X (gfx1250) — compile-verified
//
#include <hip/hip_runtime.h>
#include <math.h>

#define Bc 4
#define Tc 2048
#define Dc 512
#define Kc 4
#define NFc 4
#define NCc 32
#define Nc (Bc*Tc)          // 8192 tokens
#define Ec (2*Dc)           // 1024 (concat dim)

typedef __attribute__((ext_vector_type(2))) float v2f;
typedef __attribute__((ext_vector_type(8))) float v8f;

__device__ __forceinline__ float wave_sum(float v){
  #pragma unroll
  for (int o = 16; o > 0; o >>= 1) v += __shfl_xor(v, o, 32);
  return v;
}

__device__ __forceinline__ float gelu_exact(float h){
  return 0.5f * h * (1.0f + erff(h * 0.70710678118654752f));
}

// ---------------------------------------------------------------- k0: ||codebook_c||^2
__global__ void k0_cbnorm(const float* __restrict__ cb, float* __restrict__ cbn){
  const int wave = threadIdx.x >> 5, lane = threadIdx.x & 31;
  for (int it = 0; it < 4; ++it){
    const int c = wave + it*8;
    float a = 0.f;
    for (int d = lane; d < Dc; d += 32){ const float v = cb[(size_t)c*Dc + d]; a += v*v; }
    a = wave_sum(a);
    if (lane == 0) cbn[c] = a;
  }
}

// ---------------------------------------------------------------- k1: logic GEMM + GELU + LN + softmax(sim) K-sum
// grid = (N/8, NF), block = 256 (8 waves), 128 KB dynamic LDS.
// Each WG: 32 rows (8 tokens x K) x 512 cols for one function f.
// Each B-fragment is reused by 2 row-block WMMAs -> halves L2 weight traffic.
__global__ void __launch_bounds__(256)
k1_funclib(const float* __restrict__ x, const float* __restrict__ routed,
           const float* __restrict__ sim, const float* __restrict__ Wl,
           const float* __restrict__ bl, const float* __restrict__ lng,
           const float* __restrict__ lnb, float* __restrict__ argsum){
  extern __shared__ float ldsA[];     // 32*1024 floats, transposed: ldsA[e*32 + r]
  const int tid = threadIdx.x;
  const int f = blockIdx.y;
  const int n0 = blockIdx.x * 8;      // 8 tokens -> 32 (n,k) rows

  // stage A = combined[row, 0:1024] = [x[n] | routed[n,k]]
  for (int r = 0; r < 32; ++r){
    const int e = tid * 4;
    const int n = n0 + (r >> 2), k = r & 3;
    const float* src = (e < Dc) ? (x + (size_t)n * Dc + e)
                                : (routed + ((size_t)(n*Kc + k)) * Dc + (e - Dc));
    const float4 v = *(const float4*)src;
    ldsA[(e+0)*32 + r] = v.x; ldsA[(e+1)*32 + r] = v.y;
    ldsA[(e+2)*32 + r] = v.z; ldsA[(e+3)*32 + r] = v.w;
  }
  __syncthreads();

  const int wave = tid >> 5, lane = tid & 31;
  const int hl = lane >> 4, m16 = lane & 15;
  const int cbase = wave * 64;
  const float* Wb = Wl + (size_t)f * Ec * Dc;

  v8f acc[2][4] = {};
  for (int eo = 0; eo < Ec; eo += 32){
    if (eo + 32 < Ec)                                  // one scalar test per 8 WMMA groups
      __builtin_prefetch(Wb + (size_t)(eo + 32 + lane)*Dc + cbase, 0, 1);
    #pragma unroll
    for (int ei = 0; ei < 32; ei += 4){
      const int e0 = eo + ei;
      const int eA = (e0 + 2*hl) * 32;
      v2f a0, a1;
      a0.x = ldsA[eA      + m16]; a1.x = ldsA[eA      + 16 + m16];
      a0.y = ldsA[eA + 32 + m16]; a1.y = ldsA[eA + 48 + m16];
      #pragma unroll
      for (int t = 0; t < 4; ++t){
        const int col = cbase + t*16 + m16;
        v2f b;
        b.x = Wb[(size_t)(e0 + 2*hl    )*Dc + col];
        b.y = Wb[(size_t)(e0 + 2*hl + 1)*Dc + col];
        acc[0][t] = __builtin_amdgcn_wmma_f32_16x16x4_f32(false, a0, false, b,
                                                          (short)0, acc[0][t], false, false);
        acc[1][t] = __builtin_amdgcn_wmma_f32_16x16x4_f32(false, a1, false, b,
                                                          (short)0, acc[1][t], false, false);
      }
    }
  }
  __syncthreads();                    // safe to reuse LDS

  float* red = ldsA;                  // [0:32)=sum, [32:64)=sumsq, [64:96)=softmax(sim) wt
  if (tid < 64) red[tid] = 0.f;
  if (tid < 32){
    const int n = n0 + (tid >> 2), k = tid & 3;
    const float s0 = sim[n*Kc+0], s1 = sim[n*Kc+1], s2 = sim[n*Kc+2], s3 = sim[n*Kc+3];
    const float mx = fmaxf(fmaxf(s0,s1), fmaxf(s2,s3));
    const float e0 = expf(s0-mx), e1 = expf(s1-mx), e2 = expf(s2-mx), e3 = expf(s3-mx);
    const float inv = 1.f/(e0+e1+e2+e3);
    red[64 + tid] = (k==0 ? e0 : k==1 ? e1 : k==2 ? e2 : e3) * inv;
  }
  __syncthreads();

  // bias + exact GELU + per-row (sum,sumsq) partials, one LDS atomic per row slot
  float ps[2][8] = {}, pq[2][8] = {};
  #pragma unroll
  for (int rb = 0; rb < 2; ++rb){
    #pragma unroll
    for (int t = 0; t < 4; ++t){
      const int col = cbase + t*16 + m16;
      const float bb = bl[f*Dc + col];
      #pragma unroll
      for (int v = 0; v < 8; ++v){
        const float g = gelu_exact(acc[rb][t][v] + bb);
        acc[rb][t][v] = g;
        ps[rb][v] += g; pq[rb][v] += g*g;
      }
    }
  }
  #pragma unroll
  for (int rb = 0; rb < 2; ++rb){
    #pragma unroll
    for (int v = 0; v < 8; ++v){
      const int m = v + 8*hl + 16*rb;
      atomicAdd(&red[m], ps[rb][v]);
      atomicAdd(&red[32+m], pq[rb][v]);
    }
  }
  __syncthreads();

  // LayerNorm (per row) + weighted sum over K -> arg_summary[f, token, col]
  #pragma unroll
  for (int rb = 0; rb < 2; ++rb){
    #pragma unroll
    for (int t = 0; t < 4; ++t){
      const int col = cbase + t*16 + m16;
      const float g = lng[f*Dc + col], b = lnb[f*Dc + col];
      float o0 = 0.f, o1 = 0.f;
      #pragma unroll
      for (int v = 0; v < 8; ++v){
        const int m = v + 8*hl + 16*rb;
        const float mean = red[m] * (1.f/Dc);
        const float var  = red[32+m] * (1.f/Dc) - mean*mean;
        const float val  = (acc[rb][t][v] - mean) * rsqrtf(var + 1e-5f) * g + b;
        const float w = red[64 + m] * val;
        if (v < 4) o0 += w; else o1 += w;
      }
      const int tokA = n0 + 4*rb + 2*hl;
      argsum[((size_t)f*Nc + tokA    )*Dc + col] = o0;
      argsum[((size_t)f*Nc + tokA + 1)*Dc + col] = o1;
    }
  }
}

// ---------------------------------------------------------------- k2: fn_w = softmax(x @ W_sel + b_sel)
__global__ void k2_fnw(const float* __restrict__ x, const float* __restrict__ Wsel,
                       const float* __restrict__ bsel, float* __restrict__ fnw){
  const int wave = threadIdx.x >> 5, lane = threadIdx.x & 31;
  const int n = blockIdx.x * 8 + wave;
  float a0=0,a1=0,a2=0,a3=0;
  for (int d = lane; d < Dc; d += 32){
    const float xv = x[(size_t)n*Dc + d];
    a0 += xv * Wsel[d*NFc+0]; a1 += xv * Wsel[d*NFc+1];
    a2 += xv * Wsel[d*NFc+2]; a3 += xv * Wsel[d*NFc+3];
  }
  a0 = wave_sum(a0); a1 = wave_sum(a1); a2 = wave_sum(a2); a3 = wave_sum(a3);
  if (lane == 0){
    a0 += bsel[0]; a1 += bsel[1]; a2 += bsel[2]; a3 += bsel[3];
    const float mx = fmaxf(fmaxf(a0,a1), fmaxf(a2,a3));
    const float e0=expf(a0-mx), e1=expf(a1-mx), e2=expf(a2-mx), e3=expf(a3-mx);
    const float inv = 1.f/(e0+e1+e2+e3);
    fnw[n*NFc+0]=e0*inv; fnw[n*NFc+1]=e1*inv; fnw[n*NFc+2]=e2*inv; fnw[n*NFc+3]=e3*inv;
  }
}

// ---------------------------------------------------------------- k3: gate GEMM (x4 functions) + combine -> updated
// grid = N/16, block = 256. Each WG: 16 tokens x 512 cols; loops f, fuses sigmoid + fn_w mixing.
__global__ void __launch_bounds__(256)
k3_gate(const float* __restrict__ x, const float* __restrict__ argsum,
        const float* __restrict__ fnw, const float* __restrict__ Wg,
        const float* __restrict__ bg, float* __restrict__ upd){
  __shared__ float ldsA[16 * Ec];     // [e*16 + r]; e<512 = x (static), e>=512 = arg_summary[f]
  const int tid = threadIdx.x;
  const int n0 = blockIdx.x * 16;

  for (int r = 0; r < 16; ++r){       // stage x half once
    const int e2 = tid * 2;
    const float2 v = *(const float2*)(x + (size_t)(n0 + r)*Dc + e2);
    ldsA[(e2  )*16 + r] = v.x;
    ldsA[(e2+1)*16 + r] = v.y;
  }

  const int wave = tid >> 5, lane = tid & 31;
  const int hl = lane >> 4, m16 = lane & 15;
  const int cbase = wave * 64;

  float accP[4][8] = {}, accG[4][8] = {};

  for (int f = 0; f < NFc; ++f){
    __syncthreads();                  // previous-f LDS reads done (also covers x staging)
    for (int r = 0; r < 16; ++r){
      const int e2 = tid * 2;
      const float2 v = *(const float2*)(argsum + ((size_t)f*Nc + n0 + r)*Dc + e2);
      ldsA[(Dc + e2  )*16 + r] = v.x;
      ldsA[(Dc + e2+1)*16 + r] = v.y;
    }
    __syncthreads();

    const float* W = Wg + (size_t)f * Ec * Dc;
    v8f acc[4] = {};
    for (int eo = 0; eo < Ec; eo += 32){
      if (eo + 32 < Ec)
        __builtin_prefetch(W + (size_t)(eo + 32 + lane)*Dc + cbase, 0, 1);
      #pragma unroll
      for (int ei = 0; ei < 32; ei += 4){
        const int e0 = eo + ei;
        const int eA = (e0 + 2*hl) * 16;
        v2f a;
        a.x = ldsA[eA      + m16];
        a.y = ldsA[eA + 16 + m16];
        #pragma unroll
        for (int t = 0; t < 4; ++t){
          const int col = cbase + t*16 + m16;
          v2f b;
          b.x = W[(size_t)(e0 + 2*hl    )*Dc + col];
          b.y = W[(size_t)(e0 + 2*hl + 1)*Dc + col];
          acc[t] = __builtin_amdgcn_wmma_f32_16x16x4_f32(false, a, false, b,
                                                         (short)0, acc[t], false, false);
        }
      }
    }
    #pragma unroll
    for (int t = 0; t < 4; ++t){
      const int col = cbase + t*16 + m16;
      const float bb = bg[f*Dc + col];
      #pragma unroll
      for (int v = 0; v < 8; ++v){
        const int m = v + 8*hl;
        const float gate = 1.f / (1.f + expf(-(acc[t][v] + bb)));
        const float fw = fnw[(n0 + m)*NFc + f];
        const float as = ldsA[(Dc + col)*16 + m];   // arg_summary[f,n,col] from LDS
        accP[t][v] += fw * as;
        accG[t][v] += fw * gate;
      }
    }
  }
  #pragma unroll
  for (int t = 0; t < 4; ++t){
    const int col = cbase + t*16 + m16;
    #pragma unroll
    for (int v = 0; v < 8; ++v){
      const int m = v + 8*hl;
      upd[(size_t)(n0 + m)*Dc + col] = accP[t][v] * accG[t][v];
    }
  }
}

// ---------------------------------------------------------------- k4: Toeplitz exp kernel == bidirectional scan
__global__ void k4_scan(const float* __restrict__ upd, float* __restrict__ att,
                        const float* __restrict__ decay){
  const int idx = blockIdx.x * blockDim.x + threadIdx.x;   // 0..B*D-1
  const int b = idx / Dc, d = idx % Dc;
  const float w = expf(-decay[0]);
  const float* u = upd + (size_t)b * Tc * Dc + d;
  float* o = att + (size_t)b * Tc * Dc + d;
  float acc = 0.f;
  for (int t = 0; t < Tc; ++t){ acc = u[(size_t)t*Dc] + w*acc; o[(size_t)t*Dc] = acc; }
  acc = 0.f;
  for (int t = Tc-2; t >= 0; --t){ acc = w*(u[(size_t)(t+1)*Dc] + acc); o[(size_t)t*Dc] += acc; }
}

// ---------------------------------------------------------------- k5: LN -> VQ argmin -> LN -> out
__global__ void __launch_bounds__(256)
k5_vq(const float* __restrict__ upd, const float* __restrict__ att,
      const float* __restrict__ cb, const float* __restrict__ cbn,
      float* __restrict__ out){
  __shared__ float sWs[8], sWq[8], sdot[NCc*8];
  __shared__ float sMean, sRstd;
  __shared__ int sNear;
  const int tid = threadIdx.x, wave = tid >> 5, lane = tid & 31;
  const int row = blockIdx.x;
  const int d0 = tid*2, d1 = tid*2+1;
  const size_t base = (size_t)row * Dc;
  const float s0 = upd[base+d0] + att[base+d0];
  const float s1 = upd[base+d1] + att[base+d1];

  float ls = wave_sum(s0+s1), lq = wave_sum(s0*s0 + s1*s1);
  if (lane == 0){ sWs[wave] = ls; sWq[wave] = lq; }
  __syncthreads();
  if (tid == 0){
    float ts=0, tq=0;
    for (int w_=0; w_<8; ++w_){ ts += sWs[w_]; tq += sWq[w_]; }
    const float mean = ts*(1.f/Dc);
    sMean = mean; sRstd = rsqrtf(tq*(1.f/Dc) - mean*mean + 1e-5f);
  }
  __syncthreads();
  const float f0 = (s0 - sMean) * sRstd;
  const float f1 = (s1 - sMean) * sRstd;

  for (int c = 0; c < NCc; ++c){
    float p = f0*cb[(size_t)c*Dc + d0] + f1*cb[(size_t)c*Dc + d1];
    p = wave_sum(p);
    if (lane == 0) sdot[c*8 + wave] = p;
  }
  __syncthreads();
  if (tid < 32){
    float dt = 0.f;
    for (int w_=0; w_<8; ++w_) dt += sdot[tid*8 + w_];
    float dist = cbn[tid] - 2.f*dt;
    int ci = tid;
    for (int o = 16; o > 0; o >>= 1){
      const float od = __shfl_xor(dist, o, 32);
      const int   oi = __shfl_xor(ci,   o, 32);
      if (od < dist || (od == dist && oi < ci)){ dist = od; ci = oi; }
    }
    if (tid == 0) sNear = ci;
  }
  __syncthreads();
  const int nr = sNear;
  const float y0 = f0 + cb[(size_t)nr*Dc + d0];
  const float y1 = f1 + cb[(size_t)nr*Dc + d1];

  float ms = wave_sum(y0+y1), mq = wave_sum(y0*y0 + y1*y1);
  if (lane == 0){ sWs[wave] = ms; sWq[wave] = mq; }
  __syncthreads();
  if (tid == 0){
    float ts=0, tq=0;
    for (int w_=0; w_<8; ++w_){ ts += sWs[w_]; tq += sWq[w_]; }
    const float mean = ts*(1.f/Dc);
    sMean = mean; sRstd = rsqrtf(tq*(1.f/Dc) - mean*mean + 1e-5f);
  }
  __syncthreads();
  out[base+d0] = (y0 - sMean) * sRstd;
  out[base+d1] = (y1 - sMean) * sRstd;
}

// ----------------------------------------------------------------
extern "C" void kernel_launch(void* const* d_in, const int* in_sizes, int n_in,
                              void* d_out, int out_size, void* d_ws, size_t ws_size,
                              hipStream_t stream){
  (void)in_sizes; (void)n_in; (void)out_size; (void)ws_size;
  const float* x      = (const float*)d_in[0];
  const float* routed = (const float*)d_in[1];
  const float* sim    = (const float*)d_in[2];
  const float* Wsel   = (const float*)d_in[3];
  const float* bsel   = (const float*)d_in[4];
  const float* Wl     = (const float*)d_in[5];
  const float* bl     = (const float*)d_in[6];
  const float* lng    = (const float*)d_in[7];
  const float* lnb    = (const float*)d_in[8];
  const float* Wg     = (const float*)d_in[9];
  const float* bg     = (const float*)d_in[10];
  const float* cb     = (const float*)d_in[11];
  const float* decay  = (const float*)d_in[12];

  float* ws     = (float*)d_ws;
  float* argsum = ws;                              // NF*N*D = 16,777,216 floats
  float* fnw    = argsum + (size_t)NFc*Nc*Dc;      // N*NF   = 32,768
  float* upd    = fnw + (size_t)Nc*NFc;            // N*D    = 4,194,304
  float* att    = upd + (size_t)Nc*Dc;             // N*D    = 4,194,304
  float* cbn    = att + (size_t)Nc*Dc;             // NC     = 32

  k0_cbnorm <<<1,               256, 0,      stream>>>(cb, cbn);
  k1_funclib<<<dim3(Nc/8, NFc), 256, 131072, stream>>>(x, routed, sim, Wl, bl, lng, lnb, argsum);
  k2_fnw    <<<Nc/8,            256, 0,      stream>>>(x, Wsel, bsel, fnw);
  k3_gate   <<<Nc/16,           256, 0,      stream>>>(x, argsum, fnw, Wg, bg, upd);
  k4_scan   <<<(Bc*Dc)/256,     256, 0,      stream>>>(upd, att, decay);
  k5_vq     <<<Nc,              256, 0,      stream>>>(upd, att, cb, cbn, (float*)d_out);
}